// PrototypeMemory_48971217109505
// MI455X (gfx1250) — compile-verified
//
#include <hip/hip_runtime.h>

#define NCLS 4
#define KCH  5
#define CCH  64
#define SPATIAL (96*96*96)
#define NBATCH  2
#define NVOX    (NBATCH*SPATIAL)

typedef __attribute__((ext_vector_type(2))) float v2f;
typedef __attribute__((ext_vector_type(8))) float v8f;

// workspace layout (floats): [0..255] class sums (4x64), [256..259] counts, [260..263] sum of ||f||^2
__global__ void zero_ws_kernel(float* __restrict__ ws) {
    int i = threadIdx.x;
    if (i < NCLS * CCH + 2 * NCLS) ws[i] = 0.0f;
}

__global__ __launch_bounds__(256) void accum_kernel(
    const float* __restrict__ feat, const float* __restrict__ pred,
    const int* __restrict__ label, const unsigned char* __restrict__ is_lab,
    float* __restrict__ ws) {
    __shared__ float lsum[NCLS * CCH];
    __shared__ float lcnt[NCLS];
    __shared__ float lsq[NCLS];
    for (int i = threadIdx.x; i < NCLS * CCH; i += blockDim.x) lsum[i] = 0.0f;
    if (threadIdx.x < NCLS) { lcnt[threadIdx.x] = 0.0f; lsq[threadIdx.x] = 0.0f; }
    __syncthreads();

    const unsigned stride = gridDim.x * blockDim.x;
    for (unsigned n = blockIdx.x * blockDim.x + threadIdx.x; n < NVOX; n += stride) {
        const unsigned b = n / SPATIAL;
        const unsigned s = n - b * SPATIAL;

        // argmax + max over K=5 predictions (coalesced, non-temporal: streamed once)
        const float* pp = pred + (size_t)b * KCH * SPATIAL + s;
        float best = __builtin_nontemporal_load(pp);
        int cls = 0;
#pragma unroll
        for (int k = 1; k < KCH; ++k) {
            float v = __builtin_nontemporal_load(pp + (size_t)k * SPATIAL);
            if (v > best) { best = v; cls = k; }   // strict > keeps first max (argmax semantics)
        }
        const int lab = __builtin_nontemporal_load(label + n);
        const bool il = is_lab[b] != 0;
        const bool m = (best > 0.8f) && (cls > 0) && ((cls == lab) || !il);

        // skip the 64-channel feature read if no lane in this wave is masked
        if (__any(m)) {
            const float* fp = feat + (size_t)b * CCH * SPATIAL + s;
            const int c4 = (cls - 1) & (NCLS - 1);
            float sq = 0.0f;
#pragma unroll 4
            for (int c = 0; c < CCH; ++c) {
                float v = __builtin_nontemporal_load(fp + (size_t)c * SPATIAL);
                if (m) {
                    atomicAdd(&lsum[c4 * CCH + c], v);   // ds_add_f32, few active lanes
                    sq += v * v;
                }
            }
            if (m) {
                atomicAdd(&lcnt[c4], 1.0f);
                atomicAdd(&lsq[c4], sq);
            }
        }
    }
    __syncthreads();
    for (int i = threadIdx.x; i < NCLS * CCH; i += blockDim.x)
        if (lsum[i] != 0.0f) atomicAdd(&ws[i], lsum[i]);
    if (threadIdx.x < NCLS) {
        if (lcnt[threadIdx.x] != 0.0f) atomicAdd(&ws[256 + threadIdx.x], lcnt[threadIdx.x]);
        if (lsq[threadIdx.x]  != 0.0f) atomicAdd(&ws[260 + threadIdx.x], lsq[threadIdx.x]);
    }
}

// Single wave32. EMA update, then one chained-WMMA pass computes
//   D(16x16) = Apad(16x64) x Bpad^T-style(64x16)
// with Apad rows 0..3 = proto_new (rest 0), Bpad rows 0..3 = proto_new,
// rows 4..7 = class sums (rest 0).  Gives Gram G = D[0..3][0..3] and
// per-class dots proto_new_c . sums_c = D[c][4+c].
// Padded-zero LDS operands => unconditional ds_load_b64, uniform EXEC at WMMA.
__global__ void finalize_kernel(const float* __restrict__ ws,
                                const float* __restrict__ proto_in,
                                float* __restrict__ out) {
    __shared__ float Apad[16][CCH];
    __shared__ float Bpad[16][CCH];
    __shared__ float Dm[16][16];
    __shared__ float cntS[NCLS];
    __shared__ float sqS[NCLS];
    const int lane = threadIdx.x;

    // zero-fill pads (each lane clears 2 floats per row per array)
#pragma unroll
    for (int r = 0; r < 16; ++r) {
        Apad[r][lane] = 0.0f; Apad[r][lane + 32] = 0.0f;
        Bpad[r][lane] = 0.0f; Bpad[r][lane + 32] = 0.0f;
    }
    if (lane < NCLS) { cntS[lane] = ws[256 + lane]; sqS[lane] = ws[260 + lane]; }
    __syncthreads();

    // EMA prototype update: proto_new = present ? 0.9*p + 0.1*mean : p
#pragma unroll
    for (int c = 0; c < NCLS; ++c) {
#pragma unroll
        for (int j = 0; j < 2; ++j) {
            const int ch = lane * 2 + j;
            const float sum = ws[c * CCH + ch];
            const float cnt = cntS[c];
            const float mean = sum / fmaxf(cnt, 1.0f);
            const float p = proto_in[c * CCH + ch];
            const float pn = (cnt > 0.0f) ? 0.9f * p + 0.1f * mean : p;
            Apad[c][ch] = pn;
            Bpad[c][ch] = pn;
            Bpad[c + NCLS][ch] = sum;
            out[3 + c * CCH + ch] = pn;
        }
    }
    __syncthreads();

    // 16 chained v_wmma_f32_16x16x4_f32 over K=64 channels.
    // A 16x4 f32 layout: lanes 0-15 -> M rows, VGPR0={K0|K2 by half}, VGPR1={K1|K3}.
    // B 4x16 layout: lane&15 -> N column, VGPR0={K0|K2}, VGPR1={K1|K3}.
    v8f acc = {};
    const int half = lane >> 4;
    const int mrow = lane & 15;
#pragma unroll
    for (int kc = 0; kc < 16; ++kc) {
        const int k0 = kc * 4 + half * 2;
        v2f a, b;
        a.x = Apad[mrow][k0];
        a.y = Apad[mrow][k0 + 1];
        b.x = Bpad[mrow][k0];
        b.y = Bpad[mrow][k0 + 1];
        acc = __builtin_amdgcn_wmma_f32_16x16x4_f32(
            false, a, false, b, (short)0, acc, false, false);
    }
    // D layout: VGPR r -> M = r + 8*half, N = lane&15
#pragma unroll
    for (int r = 0; r < 8; ++r) Dm[r + 8 * half][mrow] = acc[r];
    __syncthreads();

    if (lane == 0) {
        float nvalid = 0.0f, intra = 0.0f;
        for (int c = 0; c < NCLS; ++c) {
            nvalid += cntS[c];
            // sum_masked ||f - mu_c||^2 = S2_c - 2*mu_c.sum_c + cnt_c*||mu_c||^2
            intra += sqS[c] - 2.0f * Dm[c][NCLS + c] + cntS[c] * Dm[c][c];
        }
        intra /= fmaxf(nvalid, 1.0f);
        float inter = 0.0f;
        for (int i = 0; i < NCLS; ++i)
            for (int j = i + 1; j < NCLS; ++j) {
                const float d2 = Dm[i][i] + Dm[j][j] - 2.0f * Dm[i][j];
                const float dist = sqrtf(d2 + 1e-12f);
                const float v = fmaxf(1.0f - dist, 0.0f);
                inter += v * v;
            }
        inter *= (1.0f / 6.0f);
        out[0] = 1.0f * intra + 0.1f * inter;
        out[1] = intra;
        out[2] = inter;
    }
}

extern "C" void kernel_launch(void* const* d_in, const int* in_sizes, int n_in,
                              void* d_out, int out_size, void* d_ws, size_t ws_size,
                              hipStream_t stream) {
    const float*         feat   = (const float*)d_in[0];
    const float*         pred   = (const float*)d_in[1];
    const int*           label  = (const int*)d_in[2];
    const unsigned char* is_lab = (const unsigned char*)d_in[3];
    const float*         proto  = (const float*)d_in[4];
    float* out = (float*)d_out;
    float* ws  = (float*)d_ws;

    zero_ws_kernel<<<1, 512, 0, stream>>>(ws);
    accum_kernel<<<2048, 256, 0, stream>>>(feat, pred, label, is_lab, ws);
    finalize_kernel<<<1, 32, 0, stream>>>(ws, proto, out);
}